// EpisodicMemory_64166811402570
// MI455X (gfx1250) — compile-verified
//
#include <hip/hip_runtime.h>
#include <math.h>

#define KD 512
#define VD 128
#define CAP 500000
#define NRET 32
#define NT ((CAP + 15) / 16)                       // 31250 tiles of 16 keys
#define TPB 32                                     // tiles per block (32 waves)
#define NB2 ((NT + TPB - 1) / TPB)                 // 977 sim blocks
#define NC0 (NB2 * NRET)                           // 31264 stage-0 candidates
#define NB3 32
#define CH3 ((NC0 + NB3 - 1) / NB3)                // 977 candidates per K3 block

typedef __attribute__((ext_vector_type(2))) float v2f;
typedef __attribute__((ext_vector_type(8))) float v8f;

// lanes 0-15 <-> 16-31 swap as a pure VALU op (v_permlanex16_b32, identity sel)
__device__ __forceinline__ float xorHalf(float x) {
  int i = __builtin_bit_cast(int, x);
  int r = __builtin_amdgcn_permlanex16(i, i, 0x76543210, 0xfedcba98, false, false);
  return __builtin_bit_cast(float, r);
}

__device__ __forceinline__ float waveRedSum(float p) {
#pragma unroll
  for (int off = 16; off > 0; off >>= 1) p += __shfl_xor(p, off, 32);
  return p;
}

// ---------------- K1: query MLP + LayerNorm + L2 normalize ----------------
__global__ __launch_bounds__(512) void k1_query(
    const float* __restrict__ query, const float* __restrict__ W1,
    const float* __restrict__ b1, const float* __restrict__ W2,
    const float* __restrict__ b2, const float* __restrict__ lng,
    const float* __restrict__ lnb, float* __restrict__ qOut) {
  __shared__ float qs[KD];
  __shared__ float h[KD];
  __shared__ float red[KD];
  const int tid = threadIdx.x, lane = tid & 31, w = tid >> 5;
  qs[tid] = query[tid];
  __syncthreads();
  // h = silu(q @ W1^T + b1); wave w owns rows w*32..w*32+31, coalesced float4
  for (int i = 0; i < 32; ++i) {
    const int r = (w << 5) + i;
    const float* row = W1 + (size_t)r * KD;
    float p = 0.f;
#pragma unroll
    for (int j = lane * 4; j < KD; j += 128) {
      float4 m = *(const float4*)(row + j);
      p += m.x * qs[j] + m.y * qs[j + 1] + m.z * qs[j + 2] + m.w * qs[j + 3];
    }
    p = waveRedSum(p);
    if (lane == 0) {
      float x = p + b1[r];
      h[r] = x / (1.f + expf(-x));   // silu
    }
  }
  __syncthreads();
  // h2 = h @ W2^T + b2  (store into qs, no longer needed)
  for (int i = 0; i < 32; ++i) {
    const int r = (w << 5) + i;
    const float* row = W2 + (size_t)r * KD;
    float p = 0.f;
#pragma unroll
    for (int j = lane * 4; j < KD; j += 128) {
      float4 m = *(const float4*)(row + j);
      p += m.x * h[j] + m.y * h[j + 1] + m.z * h[j + 2] + m.w * h[j + 3];
    }
    p = waveRedSum(p);
    if (lane == 0) qs[r] = p + b2[r];
  }
  __syncthreads();
  const float x = qs[tid];
  red[tid] = x;
  __syncthreads();
  for (int s = 256; s > 0; s >>= 1) { if (tid < s) red[tid] += red[tid + s]; __syncthreads(); }
  const float mu = red[0] / (float)KD;
  __syncthreads();
  const float d = x - mu;
  red[tid] = d * d;
  __syncthreads();
  for (int s = 256; s > 0; s >>= 1) { if (tid < s) red[tid] += red[tid + s]; __syncthreads(); }
  const float var = red[0] / (float)KD;
  __syncthreads();
  const float y = d * (1.f / sqrtf(var + 1e-5f)) * lng[tid] + lnb[tid];
  red[tid] = y * y;
  __syncthreads();
  for (int s = 256; s > 0; s >>= 1) { if (tid < s) red[tid] += red[tid + s]; __syncthreads(); }
  const float nrm = sqrtf(red[0]);
  qOut[tid] = y / fmaxf(nrm, 1e-12f);
}

// ------- K2: cosine sims via V_WMMA_F32_16X16X4_F32 + block top-32 --------
__global__ __launch_bounds__(1024) void k2_sims(
    const float* __restrict__ keys, const float* __restrict__ q,
    float* __restrict__ c0v, int* __restrict__ c0i) {
  __shared__ float qs[KD];
  __shared__ float sim[TPB * 16];
  __shared__ float wval[TPB];
  __shared__ int widx[TPB];
  __shared__ int bestIdx;
  const int tid = threadIdx.x, lane = tid & 31, w = tid >> 5;
  if (tid < KD) qs[tid] = q[tid];
  __syncthreads();
  const int tile = blockIdx.x * TPB + w;
  const bool hi = lane >= 16;
  if (tile < NT) {
    const int base = tile << 4;
    // lane L and L+16 cover 32 contiguous bytes of row base+(L&15) per iter
    const float* krow = keys + (size_t)(base + (lane & 15)) * KD + (hi ? 4 : 0);
    float nrm = 0.f;
    v8f acc = {0.f, 0.f, 0.f, 0.f, 0.f, 0.f, 0.f, 0.f};
    for (int k = 0; k < KD; k += 8) {
      float4 f = *(const float4*)(krow + k);
      nrm = fmaf(f.x, f.x, fmaf(f.y, f.y, fmaf(f.z, f.z, fmaf(f.w, f.w, nrm))));
      const float px = xorHalf(f.x);
      const float py = xorHalf(f.y);
      const float pz = xorHalf(f.z);
      const float pw = xorHalf(f.w);
      // A layout (16x4 f32): v0 = K0 (lanes 0-15) / K2 (16-31); v1 = K1 / K3
      v2f A0, A1, B0, B1;
      A0.x = hi ? pz : f.x;  A0.y = hi ? pw : f.y;   // cols k..k+3
      A1.x = hi ? f.z : px;  A1.y = hi ? f.w : py;   // cols k+4..k+7
      // B (4x16): every N column = same q element for its K row
      B0.x = qs[k + (hi ? 2 : 0)];     B0.y = qs[k + (hi ? 3 : 1)];
      B1.x = qs[k + 4 + (hi ? 2 : 0)]; B1.y = qs[k + 4 + (hi ? 3 : 1)];
      acc = __builtin_amdgcn_wmma_f32_16x16x4_f32(false, A0, false, B0,
                                                  (short)0, acc, false, false);
      acc = __builtin_amdgcn_wmma_f32_16x16x4_f32(false, A1, false, B1,
                                                  (short)0, acc, false, false);
    }
    nrm += xorHalf(nrm);            // lane L now has full ||key_(L&15)||^2
    const int mb = hi ? 8 : 0;      // this half-wave's D rows = keys mb..mb+7
    float s[8];
#pragma unroll
    for (int r = 0; r < 8; ++r) {
      const float n2 = __shfl(nrm, mb + r, 32);
      s[r] = acc[r] / fmaxf(sqrtf(n2), 1e-12f);
    }
    if ((lane & 15) == 0) {  // lanes 0 and 16 hold columns; write 8 sims each
#pragma unroll
      for (int r = 0; r < 8; ++r) sim[(w << 4) + mb + r] = s[r];
    }
  } else {
    if (lane < 16) sim[(w << 4) + lane] = -INFINITY;
  }
  __syncthreads();
  // block-exact top-32 over sim[512]: wave-shuffle argmax + 1-wave combine
  float v = (tid < TPB * 16) ? sim[tid] : -INFINITY;  // each thread owns one
  for (int sel = 0; sel < NRET; ++sel) {
    float bv = v;
    int bi = tid;
#pragma unroll
    for (int off = 16; off > 0; off >>= 1) {
      const float ov = __shfl_xor(bv, off, 32);
      const int oi = __shfl_xor(bi, off, 32);
      if (ov > bv) { bv = ov; bi = oi; }
    }
    if (lane == 0) { wval[w] = bv; widx[w] = bi; }
    __syncthreads();
    if (w == 0) {
      float cv = wval[lane];
      int ci = widx[lane];
#pragma unroll
      for (int off = 16; off > 0; off >>= 1) {
        const float ov = __shfl_xor(cv, off, 32);
        const int oi = __shfl_xor(ci, off, 32);
        if (ov > cv) { cv = ov; ci = oi; }
      }
      if (lane == 0) {
        c0v[blockIdx.x * NRET + sel] = cv;
        c0i[blockIdx.x * NRET + sel] = blockIdx.x * (TPB * 16) + ci;
        bestIdx = ci;
      }
    }
    __syncthreads();
    if (tid == bestIdx) v = -INFINITY;
  }
}

// ---------------- K3: reduce 31264 candidates -> 1024 ----------------
__global__ __launch_bounds__(256) void k3_reduce(
    const float* __restrict__ c0v, const int* __restrict__ c0i,
    float* __restrict__ c1v, int* __restrict__ c1i) {
  __shared__ float val[CH3];
  __shared__ int gid[CH3];
  __shared__ float rv[256];
  __shared__ int ri[256];
  const int tid = threadIdx.x;
  const int beg = blockIdx.x * CH3;
  for (int i = tid; i < CH3; i += 256) {
    const int gi = beg + i;
    val[i] = (gi < NC0) ? c0v[gi] : -INFINITY;
    gid[i] = (gi < NC0) ? c0i[gi] : 0;
  }
  __syncthreads();
  for (int sel = 0; sel < NRET; ++sel) {
    float bv = -INFINITY; int bi = 0;
    for (int i = tid; i < CH3; i += 256)
      if (val[i] > bv) { bv = val[i]; bi = i; }
    rv[tid] = bv; ri[tid] = bi;
    __syncthreads();
    for (int s = 128; s > 0; s >>= 1) {
      if (tid < s && rv[tid + s] > rv[tid]) { rv[tid] = rv[tid + s]; ri[tid] = ri[tid + s]; }
      __syncthreads();
    }
    if (tid == 0) {
      const int wi = ri[0];
      c1v[blockIdx.x * NRET + sel] = rv[0];
      c1i[blockIdx.x * NRET + sel] = gid[wi];
      val[wi] = -INFINITY;
    }
    __syncthreads();
  }
}

// -------- K4: exact global top-32, softmax, weighted value gather --------
__global__ __launch_bounds__(256) void k4_final(
    const float* __restrict__ c1v, const int* __restrict__ c1i,
    const float* __restrict__ values, float* __restrict__ out) {
  __shared__ float val[NB3 * NRET];
  __shared__ int gid[NB3 * NRET];
  __shared__ float rv[256];
  __shared__ int ri[256];
  __shared__ float topv[NRET];
  __shared__ int topi[NRET];
  __shared__ float attn[NRET];
  __shared__ float ssum;
  const int tid = threadIdx.x;
  for (int i = tid; i < NB3 * NRET; i += 256) { val[i] = c1v[i]; gid[i] = c1i[i]; }
  __syncthreads();
  for (int sel = 0; sel < NRET; ++sel) {
    float bv = -INFINITY; int bi = 0;
    for (int i = tid; i < NB3 * NRET; i += 256)
      if (val[i] > bv) { bv = val[i]; bi = i; }
    rv[tid] = bv; ri[tid] = bi;
    __syncthreads();
    for (int s = 128; s > 0; s >>= 1) {
      if (tid < s && rv[tid + s] > rv[tid]) { rv[tid] = rv[tid + s]; ri[tid] = ri[tid + s]; }
      __syncthreads();
    }
    if (tid == 0) { topv[sel] = rv[0]; topi[sel] = gid[ri[0]]; val[ri[0]] = -INFINITY; }
    __syncthreads();
  }
  if (tid < NRET) attn[tid] = expf(topv[tid] - topv[0]);  // topv[0] is the max
  __syncthreads();
  if (tid == 0) {
    float s = 0.f;
    for (int i = 0; i < NRET; ++i) s += attn[i];
    ssum = s;
  }
  __syncthreads();
  if (tid < VD) {
    float o = 0.f;
    for (int i = 0; i < NRET; ++i)
      o += (attn[i] / ssum) * values[(size_t)topi[i] * VD + tid];
    out[tid] = o;
  }
}

extern "C" void kernel_launch(void* const* d_in, const int* in_sizes, int n_in,
                              void* d_out, int out_size, void* d_ws, size_t ws_size,
                              hipStream_t stream) {
  (void)in_sizes; (void)n_in; (void)out_size; (void)ws_size;
  const float* query  = (const float*)d_in[0];
  const float* W1     = (const float*)d_in[1];
  const float* b1     = (const float*)d_in[2];
  const float* W2     = (const float*)d_in[3];
  const float* b2     = (const float*)d_in[4];
  const float* lng    = (const float*)d_in[5];
  const float* lnb    = (const float*)d_in[6];
  const float* keys   = (const float*)d_in[7];
  const float* values = (const float*)d_in[8];
  float* out = (float*)d_out;
  float* ws  = (float*)d_ws;
  // workspace layout (floats): q[512] | c0v[NC0] | c0i[NC0] | c1v[1024] | c1i[1024]
  float* q   = ws;
  float* c0v = ws + 512;
  int*   c0i = (int*)(ws + 512 + NC0);
  float* c1v = ws + 512 + 2 * NC0;
  int*   c1i = (int*)(ws + 512 + 2 * NC0 + NB3 * NRET);

  k1_query<<<dim3(1), dim3(512), 0, stream>>>(query, W1, b1, W2, b2, lng, lnb, q);
  k2_sims<<<dim3(NB2), dim3(1024), 0, stream>>>(keys, q, c0v, c0i);
  k3_reduce<<<dim3(NB3), dim3(256), 0, stream>>>(c0v, c0i, c1v, c1i);
  k4_final<<<dim3(1), dim3(256), 0, stream>>>(c1v, c1i, values, out);
}